// NAttention_3865470566917
// MI455X (gfx1250) — compile-verified
//
#include <hip/hip_runtime.h>
#include <hip/hip_bf16.h>

typedef __attribute__((ext_vector_type(2))) float v2f;
typedef __attribute__((ext_vector_type(8))) float v8f;
typedef __attribute__((ext_vector_type(4))) unsigned int v4u;
typedef __attribute__((ext_vector_type(8))) unsigned int v8u;

__device__ __forceinline__ v8f wmma_f32_16x16x4(v2f a, v2f b, v8f c) {
  // D = A(16x4) * B(4x16) + C(16x16), full f32
  return __builtin_amdgcn_wmma_f32_16x16x4_f32(false, a, false, b, (short)0, c,
                                               false, false);
}

__device__ __forceinline__ float half_max16(float x) {
  // reduce across a 16-lane half of the wave32 (xor masks keep bit4 fixed)
  x = fmaxf(x, __shfl_xor(x, 1, 32));
  x = fmaxf(x, __shfl_xor(x, 2, 32));
  x = fmaxf(x, __shfl_xor(x, 4, 32));
  x = fmaxf(x, __shfl_xor(x, 8, 32));
  return x;
}
__device__ __forceinline__ float half_sum16(float x) {
  x += __shfl_xor(x, 1, 32);
  x += __shfl_xor(x, 2, 32);
  x += __shfl_xor(x, 4, 32);
  x += __shfl_xor(x, 8, 32);
  return x;
}

// CDNA5 async global->LDS copy (GV mode), tracked by ASYNCcnt.
__device__ __forceinline__ void async_g2l_b128(const void* lds_dst,
                                               const void* gsrc) {
  const unsigned int ldsa = (unsigned int)(size_t)lds_dst;  // low 32 = LDS addr
  const unsigned long long ga = (unsigned long long)(size_t)gsrc;
  asm volatile("global_load_async_to_lds_b128 %0, %1, off" ::"v"(ldsa),
               "v"(ga)
               : "memory");
}
__device__ __forceinline__ void wait_async0() {
  asm volatile("s_wait_asynccnt 0x0" ::: "memory");
}

// ---------------------------------------------------------------------------
// Tensor Data Mover: DMA a 2D tile (rows x cols elements of f32, row stride
// `n` elements) from global memory into LDS, packed rows (cols*4 B apart).
// D# layout per CDNA5 ISA.
// ---------------------------------------------------------------------------
__device__ __forceinline__ void tdm_load_2d_f32(void* lds_dst,
                                                const void* gsrc, int n,
                                                int rows, int cols) {
  const unsigned long long ga = (unsigned long long)(size_t)gsrc;
  v4u g0;
  g0.x = 1u;                                 // count=1 (valid user descriptor)
  g0.y = (unsigned int)(size_t)lds_dst;      // lds_addr [63:32]
  g0.z = (unsigned int)ga;                   // global_addr [95:64]
  g0.w = (unsigned int)((ga >> 32) & 0x01FFFFFFull) | (2u << 30);  // type=2
  v8u g1;
  g1[0] = 0x00020000u;  // wg_mask=0, data_size=2 (4 B), no pad/iterate/barrier
  g1[1] = ((unsigned)n & 0xFFFFu) << 16;  // tensor_dim0[15:0] @ bits 63:48
  g1[2] = (((unsigned)n >> 16) & 0xFFFFu) |          // tensor_dim0[31:16]
          (((unsigned)rows & 0xFFFFu) << 16);        // tensor_dim1[15:0]
  g1[3] = (((unsigned)rows >> 16) & 0xFFFFu) |       // tensor_dim1[31:16]
          (((unsigned)cols & 0xFFFFu) << 16);        // tile_dim0
  g1[4] = (unsigned)rows & 0xFFFFu;                  // tile_dim1 (tile_dim2=0)
  g1[5] = (unsigned)n;                               // tensor_dim0_stride low
  g1[6] = 0u;                                        // stride hi / dim1_stride
  g1[7] = 0u;
  asm volatile("tensor_load_to_lds %0, %1" ::"s"(g0), "s"(g1) : "memory");
}

// ---------------------------------------------------------------------------
// WMMA f32 GEMM: C[M,N] = A[M,K] @ B[K,N] + bias[N]; cols < scale_cols get
// *scale. Block tile (WM*32) x (WN*32); each wave owns a 32x32 sub-tile.
// Whole KxBN weight panel DMA'd once by the TDM; A K-tiles double-buffered
// through GLOBAL_LOAD_ASYNC_TO_LDS_B128 so loads overlap the WMMA chain.
// ---------------------------------------------------------------------------
template <int WM, int WN, int KMAX>
__global__ __launch_bounds__(32 * WM * WN) void gemm_wmma(
    const float* __restrict__ A, const float* __restrict__ B,
    const float* __restrict__ bias, float* __restrict__ C, int M, int N,
    int Kd, float scale, int scale_cols) {
  constexpr int BM = WM * 32, BN = WN * 32, BK = 32, NT = 32 * WM * WN;
  // +4 float pad: keeps rows 16B-aligned for B128 async writes AND maps the
  // 16 strided A-side reads onto 16 distinct LDS banks (36*i mod 64 distinct).
  __shared__ float sA[2][BM][BK + 4];
  __shared__ float sB[KMAX][BN];  // full K x BN weight panel (TDM-resident)
  const int tid = threadIdx.x;
  const int wave = tid >> 5, lane = tid & 31, lo = lane & 15, hi = lane >> 4;
  const int wm = wave / WN, wn = wave % WN;
  const int m0 = blockIdx.y * BM, n0 = blockIdx.x * BN;

  // One DMA per workgroup: B panel Kd x BN, row stride N (TDM ignores EXEC,
  // so only wave 0 issues; it waits TENSORcnt, then the barrier publishes).
  if (wave == 0) {
    tdm_load_2d_f32(&sB[0][0], B + n0, N, Kd, BN);
    __builtin_amdgcn_s_wait_tensorcnt(0);
  }

  auto issueA = [&](int kc, int buf) {
    for (int i = tid; i < BM * 8; i += NT) {
      const int r = i >> 3, c = (i & 7) * 4;
      async_g2l_b128(&sA[buf][r][c], &A[(size_t)(m0 + r) * Kd + kc + c]);
    }
  };
  issueA(0, 0);

  v8f acc[2][2] = {};
  const int T = Kd / BK;
  for (int t = 0; t < T; ++t) {
    wait_async0();    // A tile t landed (issued one full compute step ago)
    __syncthreads();  // publish tile t; everyone done reading buf (t+1)&1
    if (t + 1 < T) issueA((t + 1) * BK, (t + 1) & 1);  // overlap next tile
    const float(*curA)[BK + 4] = sA[t & 1];
    const int kb = t * BK;
#pragma unroll
    for (int sk = 0; sk < 8; ++sk) {
      const int k = sk * 4 + 2 * hi;  // A: K in {VGPR j, lane half}
      v2f a[2], b[2];
#pragma unroll
      for (int mi = 0; mi < 2; ++mi) {
        const int mr = wm * 32 + mi * 16 + lo;
        a[mi].x = curA[mr][k];
        a[mi].y = curA[mr][k + 1];
      }
#pragma unroll
      for (int ni = 0; ni < 2; ++ni) {
        const int nc = wn * 32 + ni * 16 + lo;
        b[ni].x = sB[kb + k][nc];
        b[ni].y = sB[kb + k + 1][nc];
      }
#pragma unroll
      for (int mi = 0; mi < 2; ++mi)
#pragma unroll
        for (int ni = 0; ni < 2; ++ni)
          acc[mi][ni] = wmma_f32_16x16x4(a[mi], b[ni], acc[mi][ni]);
    }
  }
#pragma unroll
  for (int mi = 0; mi < 2; ++mi)
#pragma unroll
    for (int ni = 0; ni < 2; ++ni)
#pragma unroll
      for (int r = 0; r < 8; ++r) {
        const int row = m0 + wm * 32 + mi * 16 + r + 8 * hi;
        const int col = n0 + wn * 32 + ni * 16 + lo;
        float v = acc[mi][ni][r] + bias[col];
        if (col < scale_cols) v *= scale;
        C[(size_t)row * N + col] = v;
      }
}

// ---------------------------------------------------------------------------
// Neighborhood attention, flash-style. One wave32 per (b, row, 16-px tile,
// head). qkv layout: [b, pix, {q|k|v} x head x 32]; q pre-scaled.
// ---------------------------------------------------------------------------
__global__ __launch_bounds__(32) void natt_kernel(const float* __restrict__ qkv,
                                                  const float* __restrict__ rpb,
                                                  float* __restrict__ out) {
  constexpr int Hh = 192, Ww = 192, HEADS = 6, HD = 32, DIM = 192, C3 = 576;
  constexpr int L = Hh * Ww;
  int id = blockIdx.x;
  const int head = id % HEADS;
  id /= HEADS;
  const int wt = id % (Ww / 16);
  id /= (Ww / 16);
  const int hr = id % Hh;
  const int b = id / Hh;
  const int w0 = wt * 16;
  const int lane = threadIdx.x & 31, lo = lane & 15, hi = lane >> 4;

  const size_t img = (size_t)b * L;

  // Q tile in WMMA A layout: 8 chunks of K=4 over head_dim=32
  v2f qa[8];
  {
    const float* qp = qkv + (img + (size_t)hr * Ww + w0 + lo) * C3 + head * HD;
#pragma unroll
    for (int c = 0; c < 8; ++c) {
      const int d0 = 4 * c + 2 * hi;
      qa[c].x = qp[d0];
      qa[c].y = qp[d0 + 1];
    }
  }
  const int rbase = min(max(hr - 3, 0), Hh - 7);
  const int cbase = min(max(w0 - 3, 0), Ww - 7);
  int swv[8];  // per-pixel window start for the rows this lane half owns
#pragma unroll
  for (int r = 0; r < 8; ++r) swv[r] = min(max(w0 + r + 8 * hi - 3, 0), Ww - 7);

  float mrun[8], lrun[8];
#pragma unroll
  for (int r = 0; r < 8; ++r) {
    mrun[r] = -1e30f;
    lrun[r] = 0.0f;
  }
  v8f oacc[2] = {};  // 16 px x 32 dims accumulator (two 16x16 C tiles)

  __shared__ float lds_p[16][17];  // C-layout -> A-layout staging for P
  const float* rpb_h = rpb + head * 13 * 13;

  for (int ri = 0; ri < 7; ++ri) {
    const int row = rbase + ri;
    const int bih = row - hr + 6;
    const float* kbase = qkv + (img + (size_t)row * Ww) * C3 + DIM + head * HD;
    const float* vbase = kbase + DIM;
    for (int ct = 0; ct < 2; ++ct) {
      // ---- logits tile: Q(16x32) @ K^T(32x16) via 8 chained f32 WMMAs ----
      const int ncol = cbase + ct * 16 + lo;  // lane carries N = neighbor col
      const float* kp = kbase + (size_t)min(ncol, Ww - 1) * C3;
      v8f lg = {};
#pragma unroll
      for (int c = 0; c < 8; ++c) {
        const int d0 = 4 * c + 2 * hi;
        v2f kb;
        kb.x = kp[d0];
        kb.y = kp[d0 + 1];
        lg = wmma_f32_16x16x4(qa[c], kb, lg);
      }
      // ---- bias + mask + online softmax (per output row) ----
      float pst[8];
#pragma unroll
      for (int r = 0; r < 8; ++r) {
        const int m = r + 8 * hi;  // pixel row this VGPR holds in this half
        const bool inw = (ncol >= swv[r]) & (ncol <= swv[r] + 6);
        const float lv =
            inw ? (lg[r] + rpb_h[bih * 13 + (ncol - (w0 + m) + 6)]) : -1e30f;
        const float rmax = half_max16(lv);
        const float mnew = fmaxf(mrun[r], rmax);
        const float sc = __expf(mrun[r] - mnew);
        const float p = inw ? __expf(lv - mnew) : 0.0f;
        lrun[r] = lrun[r] * sc + half_sum16(p);
        mrun[r] = mnew;
        oacc[0][r] *= sc;
        oacc[1][r] *= sc;
        pst[r] = p;
      }
      // ---- transpose P via LDS into A layout ----
      __syncthreads();
#pragma unroll
      for (int r = 0; r < 8; ++r) lds_p[r + 8 * hi][lo] = pst[r];
      __syncthreads();
      // ---- O += P(16x16) @ V(16x32) via 8 chained f32 WMMAs ----
#pragma unroll
      for (int c2 = 0; c2 < 4; ++c2) {
        const int nb = 4 * c2 + 2 * hi;
        v2f pa;
        pa.x = lds_p[lo][nb];
        pa.y = lds_p[lo][nb + 1];
        const int col0 = min(cbase + ct * 16 + nb, Ww - 1);
        const int col1 = min(cbase + ct * 16 + nb + 1, Ww - 1);
        const float* vp0 = vbase + (size_t)col0 * C3;
        const float* vp1 = vbase + (size_t)col1 * C3;
#pragma unroll
        for (int tt = 0; tt < 2; ++tt) {
          const int d = tt * 16 + lo;
          v2f vb;
          vb.x = vp0[d];
          vb.y = vp1[d];
          oacc[tt] = wmma_f32_16x16x4(pa, vb, oacc[tt]);
        }
      }
    }
  }
  // ---- normalize and store [b, pix, head*32 + d] ----
#pragma unroll
  for (int r = 0; r < 8; ++r) {
    const float inv = 1.0f / lrun[r];
    const size_t o =
        (img + (size_t)hr * Ww + w0 + r + 8 * hi) * DIM + head * HD;
    out[o + lo] = oacc[0][r] * inv;
    out[o + 16 + lo] = oacc[1][r] * inv;
  }
}

extern "C" void kernel_launch(void* const* d_in, const int* in_sizes, int n_in,
                              void* d_out, int out_size, void* d_ws,
                              size_t ws_size, hipStream_t stream) {
  const float* x = (const float*)d_in[0];
  const float* w_qkv = (const float*)d_in[1];
  const float* b_qkv = (const float*)d_in[2];
  const float* rpb = (const float*)d_in[3];
  const float* w_out = (const float*)d_in[4];
  const float* b_out = (const float*)d_in[5];

  constexpr int Bb = 4, Hh = 192, Ww = 192, C_IN = 288, DIM = 192;
  constexpr int HEADS = 6;
  constexpr int L = Hh * Ww;   // 36864
  constexpr int M = Bb * L;    // 147456
  constexpr int C3 = 3 * DIM;  // 576

  float* qkv_ws = (float*)d_ws;              // M*576 floats (~340 MB)
  float* attn_ws = qkv_ws + (size_t)M * C3;  // M*192 floats (~113 MB)

  // 1) fused QKV projection; q columns (<192) scaled by head_dim^-1/2
  gemm_wmma<2, 2, 288><<<dim3(C3 / 64, M / 64), 128, 0, stream>>>(
      x, w_qkv, b_qkv, qkv_ws, M, C3, C_IN, 0.17677669529663687f, DIM);

  // 2) flash-style neighborhood attention: one wave per (b,row,16px,head)
  natt_kernel<<<Bb * Hh * (Ww / 16) * HEADS, 32, 0, stream>>>(qkv_ws, rpb,
                                                              attn_ws);

  // 3) output projection to DIM/2 = 96
  gemm_wmma<4, 1, 192><<<dim3(96 / 32, M / 128), 128, 0, stream>>>(
      attn_ws, w_out, b_out, (float*)d_out, M, 96, DIM, 1.0f, 0);
}